// Topk_pool_3401614098591
// MI455X (gfx1250) — compile-verified
//
#include <hip/hip_runtime.h>
#include <hip/hip_bf16.h>

// ---------------------------------------------------------------------------
// TopKPooling on MI455X (gfx1250, wave32).
// Pipeline: WMMA-scored GEMV -> degree/GCN edge scatters -> per-graph bitonic
// top-k -> gather * tanh(score) -> edge relabel/filter.
// ---------------------------------------------------------------------------

#define N_GRAPHS        64
#define NODES_PER_GRAPH 1024
#define N_NODES         (N_GRAPHS * NODES_PER_GRAPH)   // 65536
#define AVG_DEGREE      32
#define N_EDGES         (N_NODES * AVG_DEGREE)         // 2097152
#define D_FEAT          256
#define ALPHA           0.5f
#define KSEL            512                            // kept nodes per graph
#define N_OUT           (N_GRAPHS * KSEL)              // 32768
#define KSTEPS          (D_FEAT / 4)                   // 64 WMMA K-steps

typedef __attribute__((ext_vector_type(2)))  float    v2f;
typedef __attribute__((ext_vector_type(8)))  float    v8f;
typedef __attribute__((ext_vector_type(16))) _Float16 v16h;

// ---------------------------------------------------------------------------
// 0) init: deg=0, gcn=0, node_map=-1
// ---------------------------------------------------------------------------
__global__ __launch_bounds__(256) void init_kernel(float* __restrict__ deg,
                                                   float* __restrict__ gcn,
                                                   int* __restrict__ node_map) {
    int i = blockIdx.x * 256 + threadIdx.x;
    if (i < N_NODES) {
        deg[i] = 0.0f;
        gcn[i] = 0.0f;
        node_map[i] = -1;
    }
}

// ---------------------------------------------------------------------------
// 1) Fused GEMV via fp32 WMMA: xw1 = x@w1, xwg = x@wg.
//    One wave handles 16 nodes.  B tile (4x16 per step) holds w1 in column 0
//    and wg in column 1; it is pre-swizzled into an LDS table indexed by
//    (step, lane) so the hot loop is branch-free:
//        global_load_b64 (A) + ds_load_b64 (B) + v_wmma_f32_16x16x4_f32.
// ---------------------------------------------------------------------------
__global__ __launch_bounds__(256) void score_matvec_wmma(
    const float* __restrict__ x, const float* __restrict__ w1,
    const float* __restrict__ wg, float* __restrict__ xw1,
    float* __restrict__ xwg) {
    // Pre-swizzled B operand: sB[step*32 + lane] = per-lane v2f for that step.
    __shared__ v2f sB[KSTEPS * 32];                    // 16 KB

    int tid = threadIdx.x;
    for (int e = tid; e < KSTEPS * 32; e += 256) {
        int step = e >> 5;
        int ln   = e & 31;
        int mm   = ln & 15;                            // B column
        int kh   = (ln < 16) ? 0 : 2;                  // K split across halves
        int k    = step * 4 + kh;
        v2f b; b.x = 0.0f; b.y = 0.0f;
        if (mm == 0)      { b.x = w1[k]; b.y = w1[k + 1]; }
        else if (mm == 1) { b.x = wg[k]; b.y = wg[k + 1]; }
        sB[e] = b;
    }
    __syncthreads();

    int wave = tid >> 5;
    int lane = tid & 31;
    int m    = lane & 15;                              // A row-in-tile
    int khalf = (lane < 16) ? 0 : 2;                   // A: lanes 16-31 -> K=2,3
    int node_base = (blockIdx.x * 8 + wave) * 16;
    const float* __restrict__ arow = x + (size_t)(node_base + m) * D_FEAT;

    __builtin_prefetch(arow, 0, 1);                    // global_prefetch_b8

    v8f c = {};
#pragma unroll 8
    for (int step = 0; step < KSTEPS; ++step) {
        int k = step * 4;
        v2f a;
        a.x = arow[k + khalf];
        a.y = arow[k + khalf + 1];
        v2f b = sB[step * 32 + lane];
        c = __builtin_amdgcn_wmma_f32_16x16x4_f32(
                false, a, false, b, (short)0, c, false, false);
    }

    // D layout: VGPR j -> M = j + (lane<16 ? 0 : 8), N = lane&15.
    // Column 0 (lanes 0,16) = x@w1 ; column 1 (lanes 1,17) = x@wg.
    if (m < 2) {
        float* dst = (m == 0) ? xw1 : xwg;
        int rbase = node_base + ((lane < 16) ? 0 : 8);
#pragma unroll
        for (int j = 0; j < 8; ++j) dst[rbase + j] = c[j];
    }
}

// ---------------------------------------------------------------------------
// 2) degree scatter: deg[col] += (row != col)
// ---------------------------------------------------------------------------
__global__ __launch_bounds__(256) void deg_kernel(const int* __restrict__ ei,
                                                  float* __restrict__ deg) {
    int e = blockIdx.x * 256 + threadIdx.x;
    if (e < N_EDGES) {
        int row = ei[e];
        int col = ei[N_EDGES + e];
        if (row != col) atomicAdd(&deg[col], 1.0f);
    }
}

// 3) dis[i] = deg>0 ? rsqrt(max(deg,1)) : 0
__global__ __launch_bounds__(256) void dis_kernel(const float* __restrict__ deg,
                                                  float* __restrict__ dis) {
    int i = blockIdx.x * 256 + threadIdx.x;
    if (i < N_NODES) {
        float d = deg[i];
        dis[i] = (d > 0.0f) ? rsqrtf(fmaxf(d, 1.0f)) : 0.0f;
    }
}

// 4) gcn[col] += dis[row]*dis[col]*xwg[row]  (self-loops removed)
__global__ __launch_bounds__(256) void gcn_kernel(const int* __restrict__ ei,
                                                  const float* __restrict__ dis,
                                                  const float* __restrict__ xwg,
                                                  float* __restrict__ gcn) {
    int e = blockIdx.x * 256 + threadIdx.x;
    if (e < N_EDGES) {
        int row = ei[e];
        int col = ei[N_EDGES + e];
        if (row != col) {
            float nrm = dis[row] * dis[col];
            atomicAdd(&gcn[col], nrm * xwg[row]);
        }
    }
}

// 5) score = ALPHA*(xw1 + b1) + (1-ALPHA)*(gcn + bg)
__global__ __launch_bounds__(256) void score_kernel(const float* __restrict__ xw1,
                                                    const float* __restrict__ gcn,
                                                    const float* __restrict__ b1,
                                                    const float* __restrict__ bg,
                                                    float* __restrict__ score) {
    int i = blockIdx.x * 256 + threadIdx.x;
    if (i < N_NODES) {
        score[i] = ALPHA * (xw1[i] + b1[0]) + (1.0f - ALPHA) * (gcn[i] + bg[0]);
    }
}

// ---------------------------------------------------------------------------
// 6) per-graph top-K: full bitonic sort of 1024 (score desc, idx asc) in LDS,
//    keep first 512.  One block (32 waves) per graph.
// ---------------------------------------------------------------------------
__global__ __launch_bounds__(1024) void topk_kernel(const float* __restrict__ score,
                                                    int* __restrict__ perm,
                                                    float* __restrict__ score_perm) {
    __shared__ float skey[NODES_PER_GRAPH];
    __shared__ int   sidx[NODES_PER_GRAPH];
    int g = blockIdx.x;
    int tid = threadIdx.x;
    skey[tid] = score[g * NODES_PER_GRAPH + tid];
    sidx[tid] = tid;
    __syncthreads();

    for (int k2 = 2; k2 <= NODES_PER_GRAPH; k2 <<= 1) {
        for (int j = k2 >> 1; j > 0; j >>= 1) {
            int ixj = tid ^ j;
            if (ixj > tid) {
                float ka = skey[tid], kb = skey[ixj];
                int ia = sidx[tid], ib = sidx[ixj];
                bool desc = ((tid & k2) == 0);
                bool a_better = (ka > kb) || (ka == kb && ia < ib);
                bool do_swap = desc ? (!a_better) : a_better;
                if (do_swap) {
                    skey[tid] = kb; skey[ixj] = ka;
                    sidx[tid] = ib; sidx[ixj] = ia;
                }
            }
            __syncthreads();
        }
    }

    if (tid < KSEL) {
        perm[g * KSEL + tid]       = g * NODES_PER_GRAPH + sidx[tid];
        score_perm[g * KSEL + tid] = skey[tid];
    }
}

// 7) node_map[perm[j]] = j
__global__ __launch_bounds__(256) void scatter_map_kernel(const int* __restrict__ perm,
                                                          int* __restrict__ node_map) {
    int j = blockIdx.x * 256 + threadIdx.x;
    if (j < N_OUT) node_map[perm[j]] = j;
}

// ---------------------------------------------------------------------------
// 8) gather: x_new[j,:] = x[perm[j],:] * tanh(score[perm[j]]) ; batch_new.
//    float4 traffic; 4 rows per 256-thread block (64 lanes/row).
// ---------------------------------------------------------------------------
__global__ __launch_bounds__(256) void gather_kernel(const float* __restrict__ x,
                                                     const int* __restrict__ perm,
                                                     const float* __restrict__ score_perm,
                                                     float* __restrict__ out_x,
                                                     float* __restrict__ out_batch) {
    int tid = threadIdx.x;
    int r = blockIdx.x * 4 + (tid >> 6);
    int c4 = tid & 63;
    if (r < N_OUT) {
        int p = perm[r];
        float t = tanhf(score_perm[r]);
        const float4* __restrict__ src =
            (const float4*)(x + (size_t)p * D_FEAT);
        float4* __restrict__ dst = (float4*)(out_x + (size_t)r * D_FEAT);
        float4 v = src[c4];
        v.x *= t; v.y *= t; v.z *= t; v.w *= t;
        dst[c4] = v;
        if (c4 == 0) out_batch[r] = (float)(p >> 10);  // p / NODES_PER_GRAPH
    }
}

// 9) filter_adj: relabel edges, -1 when either endpoint dropped (as floats)
__global__ __launch_bounds__(256) void edge_filter_kernel(const int* __restrict__ ei,
                                                          const int* __restrict__ node_map,
                                                          float* __restrict__ out_ei) {
    int e = blockIdx.x * 256 + threadIdx.x;
    if (e < N_EDGES) {
        int er = node_map[ei[e]];
        int ec = node_map[ei[N_EDGES + e]];
        bool valid = (er >= 0) && (ec >= 0);
        out_ei[e]           = valid ? (float)er : -1.0f;
        out_ei[N_EDGES + e] = valid ? (float)ec : -1.0f;
    }
}

// ---------------------------------------------------------------------------
extern "C" void kernel_launch(void* const* d_in, const int* in_sizes, int n_in,
                              void* d_out, int out_size, void* d_ws, size_t ws_size,
                              hipStream_t stream) {
    const float* x    = (const float*)d_in[0];
    const int*   ei   = (const int*)d_in[1];
    // d_in[2] = batch (unused: graph id recomputed from perm/1024)
    const float* w1   = (const float*)d_in[3];
    const float* b1   = (const float*)d_in[4];
    const float* wg   = (const float*)d_in[5];
    const float* bg   = (const float*)d_in[6];
    (void)in_sizes; (void)n_in; (void)out_size; (void)ws_size;

    // workspace carve-up
    float* ws        = (float*)d_ws;
    float* xw1       = ws;
    float* xwg       = ws + 1 * N_NODES;
    float* deg       = ws + 2 * N_NODES;
    float* dis       = ws + 3 * N_NODES;
    float* gcn       = ws + 4 * N_NODES;
    float* score     = ws + 5 * N_NODES;
    float* score_prm = ws + 6 * N_NODES;                    // N_OUT
    int*   perm      = (int*)(ws + 6 * N_NODES + N_OUT);    // N_OUT
    int*   node_map  = (int*)(ws + 6 * N_NODES + 2 * N_OUT);// N_NODES

    // output carve-up (flattened tuple, all as float)
    float* out_x     = (float*)d_out;
    float* out_ei    = out_x + (size_t)N_OUT * D_FEAT;
    float* out_batch = out_ei + (size_t)2 * N_EDGES;

    const int NB_N = N_NODES / 256;   // 256
    const int NB_E = N_EDGES / 256;   // 8192

    init_kernel<<<NB_N, 256, 0, stream>>>(deg, gcn, node_map);
    score_matvec_wmma<<<N_NODES / 128, 256, 0, stream>>>(x, w1, wg, xw1, xwg);
    deg_kernel<<<NB_E, 256, 0, stream>>>(ei, deg);
    dis_kernel<<<NB_N, 256, 0, stream>>>(deg, dis);
    gcn_kernel<<<NB_E, 256, 0, stream>>>(ei, dis, xwg, gcn);
    score_kernel<<<NB_N, 256, 0, stream>>>(xw1, gcn, b1, bg, score);
    topk_kernel<<<N_GRAPHS, 1024, 0, stream>>>(score, perm, score_prm);
    scatter_map_kernel<<<N_OUT / 256, 256, 0, stream>>>(perm, node_map);
    gather_kernel<<<N_OUT / 4, 256, 0, stream>>>(x, perm, score_prm, out_x, out_batch);
    edge_filter_kernel<<<NB_E, 256, 0, stream>>>(ei, node_map, out_ei);
}